// Downsampler_31860067401728
// MI455X (gfx1250) — compile-verified
//
#include <hip/hip_runtime.h>

// ---------------- problem constants (from reference) ----------------
#define NB      8
#define NC      3
#define H_      512
#define W_      512
#define HO_     128
#define WO_     128
#define PAD     6
#define HP_     524           // H_ + 2*PAD
#define WP_     524
#define NTAPS   169           // 13*13
#define PIX     (HO_ * WO_)   // 16384
#define TPB     256
// async pipeline: 4 buffers x 4 taps x 3 tensors x 256 threads
#define GTAPS   4
#define NBUF    4
#define NSTAGES 43            // ceil(169/4)

// 1 = issue async loads via inline asm (opaque to compiler: our counted
//     s_wait_asynccnt 36 is the only drain -> true 4-stage pipeline)
// 0 = use the clang builtin (known to compile; compiler may add
//     conservative full drains before the LDS reads)
#define ASYNC_VIA_ASM 1

struct F3 { float x, y, z; };

__device__ __forceinline__ int reflect512(int q) {
  int r = q < 0 ? -q : q;          // left reflect (numpy 'reflect', no edge dup)
  return r > 511 ? 1022 - r : r;   // right reflect
}

// one lane: 4 bytes global -> LDS, tracked on ASYNCcnt
__device__ __forceinline__ void async_copy_b32(const float* gsrc, float* ldst) {
#if ASYNC_VIA_ASM
  unsigned loff = (unsigned)(unsigned long long)
      (__attribute__((address_space(3))) int*)ldst;
  asm volatile("global_load_async_to_lds_b32 %0, %1, off"
               :: "v"(loff), "v"(gsrc) : "memory");
#else
  __builtin_amdgcn_global_load_async_to_lds_b32(
      (__attribute__((address_space(1))) int*)(const_cast<float*>(gsrc)),
      (__attribute__((address_space(3))) int*)ldst,
      0, 0);
#endif
}

// --------------------------- main kernel ---------------------------
template <bool NHWC>
__global__ __launch_bounds__(TPB) void downsample_kernel(
    const float* __restrict__ img,   // NCHW original (fallback path)
    const float* __restrict__ kern,
    const float* __restrict__ offh,
    const float* __restrict__ offv,
    const float* __restrict__ wsp,   // padded NHWC (fast path)
    float* __restrict__ out) {
  __shared__ float lds[NBUF][GTAPS][3][TPB];

  const int tid = threadIdx.x;
  const int pix = blockIdx.x * TPB + tid;   // block never crosses an image
  const int b   = pix >> 14;                // / PIX
  const int p   = pix & (PIX - 1);
  const int ho  = p >> 7;
  const int wo  = p & (WO_ - 1);

  const size_t tapBase = (size_t)b * NTAPS * PIX + (size_t)p;

  auto issue = [&](int s) {
    float (*buf)[3][TPB] = lds[s & (NBUF - 1)];
#pragma unroll
    for (int g = 0; g < GTAPS; ++g) {
      int kk = s * GTAPS + g;
      if (kk > NTAPS - 1) kk = NTAPS - 1;   // phantom tail: uniform accounting
      const size_t off = tapBase + (size_t)kk * PIX;
      async_copy_b32(kern + off, &buf[g][0][tid]);
      async_copy_b32(offh + off, &buf[g][1][tid]);
      async_copy_b32(offv + off, &buf[g][2][tid]);
    }
  };

  // prologue: fill the 4-stage pipeline (48 outstanding async loads max)
#pragma unroll
  for (int s = 0; s < NBUF; ++s) issue(s);

  const float pxb = (float)wo * 4.0f + 1.5f;  // (wo+0.5)*4 - 0.5
  const float pyb = (float)ho * 4.0f + 1.5f;

  float s0 = 0.f, s1 = 0.f, s2 = 0.f;

  for (int s = 0; s < NSTAGES; ++s) {
    // async loads complete in order: <=36 outstanding => oldest stage (12) done
    asm volatile("s_wait_asynccnt 36" ::: "memory");
    float (*buf)[3][TPB] = lds[s & (NBUF - 1)];
#pragma unroll
    for (int g = 0; g < GTAPS; ++g) {
      const int kk = s * GTAPS + g;
      if (kk < NTAPS) {
        const float kw = buf[g][0][tid];
        const float oh = buf[g][1][tid];
        const float ov = buf[g][2][tid];
        const int   ky = kk / 13;
        const int   kx = kk - ky * 13;
        const float px = pxb + (float)kx + oh * 4.0f;
        const float py = pyb + (float)ky + ov * 4.0f;
        const float fx = floorf(px), fy = floorf(py);
        const float a  = px - fx;
        const float bt = py - fy;
        int xL = (int)fx; xL = xL < 0 ? 0 : (xL > WP_ - 1 ? WP_ - 1 : xL);
        int yT = (int)fy; yT = yT < 0 ? 0 : (yT > HP_ - 1 ? HP_ - 1 : yT);
        const int xR = (xL + 1 > WP_ - 1) ? WP_ - 1 : xL + 1;
        const int yB = (yT + 1 > HP_ - 1) ? HP_ - 1 : yT + 1;
        const float w00 = (1.f - a) * (1.f - bt);
        const float w10 = a * (1.f - bt);
        const float w01 = (1.f - a) * bt;
        const float w11 = a * bt;
        if (NHWC) {
          const F3* r0 = (const F3*)wsp + (size_t)(b * HP_ + yT) * WP_;
          const F3* r1 = (const F3*)wsp + (size_t)(b * HP_ + yB) * WP_;
          const F3 tl = r0[xL], tr = r0[xR], bl = r1[xL], br = r1[xR];
          s0 += kw * (w00 * tl.x + w10 * tr.x + w01 * bl.x + w11 * br.x);
          s1 += kw * (w00 * tl.y + w10 * tr.y + w01 * bl.y + w11 * br.y);
          s2 += kw * (w00 * tl.z + w10 * tr.z + w01 * bl.z + w11 * br.z);
        } else {
          const int x0 = reflect512(xL - PAD), x1 = reflect512(xR - PAD);
          const int y0 = reflect512(yT - PAD), y1 = reflect512(yB - PAD);
#pragma unroll
          for (int c = 0; c < 3; ++c) {
            const float* plane = img + ((size_t)(b * 3 + c) * H_) * W_;
            const float tl = plane[(size_t)y0 * W_ + x0];
            const float tr = plane[(size_t)y0 * W_ + x1];
            const float bl = plane[(size_t)y1 * W_ + x0];
            const float br = plane[(size_t)y1 * W_ + x1];
            const float v = kw * (w00 * tl + w10 * tr + w01 * bl + w11 * br);
            if (c == 0) s0 += v; else if (c == 1) s1 += v; else s2 += v;
          }
        }
      }
    }
    // drain our DS reads before async writes may reuse this buffer (WAR)
    asm volatile("s_wait_dscnt 0" ::: "memory");
    issue(s + NBUF);
  }

  out[(size_t)(b * 3 + 0) * PIX + p] = s0;
  out[(size_t)(b * 3 + 1) * PIX + p] = s1;
  out[(size_t)(b * 3 + 2) * PIX + p] = s2;
}

// force emission order: main kernels first (so disasm snippet shows them)
template __global__ void downsample_kernel<true>(
    const float*, const float*, const float*, const float*, const float*, float*);
template __global__ void downsample_kernel<false>(
    const float*, const float*, const float*, const float*, const float*, float*);

// ------------- pre-pass: reflect-pad + NCHW -> padded NHWC -------------
__global__ void pad_nhwc_kernel(const float* __restrict__ img,
                                float* __restrict__ wsp) {
  int t = blockIdx.x * blockDim.x + threadIdx.x;
  const int total = NB * HP_ * WP_;
  if (t >= total) return;
  int xp = t % WP_;
  int r  = t / WP_;
  int yp = r % HP_;
  int b  = r / HP_;
  int x = reflect512(xp - PAD);
  int y = reflect512(yp - PAD);
  size_t src = (((size_t)b * NC) * H_ + y) * W_ + x;
  float v0 = img[src];
  float v1 = img[src + (size_t)H_ * W_];
  float v2 = img[src + 2ull * H_ * W_];
  size_t d = (size_t)t * 3;
  wsp[d + 0] = v0;
  wsp[d + 1] = v1;
  wsp[d + 2] = v2;
}

// --------------------------- launcher ---------------------------
extern "C" void kernel_launch(void* const* d_in, const int* in_sizes, int n_in,
                              void* d_out, int out_size, void* d_ws, size_t ws_size,
                              hipStream_t stream) {
  const float* img  = (const float*)d_in[0];
  const float* kern = (const float*)d_in[1];
  const float* offh = (const float*)d_in[2];
  const float* offv = (const float*)d_in[3];
  float* out = (float*)d_out;

  const size_t need = (size_t)NB * HP_ * WP_ * 3 * sizeof(float);
  const int nblocks = (NB * PIX) / TPB;  // 512

  if (d_ws && ws_size >= need) {
    float* wsp = (float*)d_ws;
    const int padBlocks = (NB * HP_ * WP_ + TPB - 1) / TPB;
    pad_nhwc_kernel<<<padBlocks, TPB, 0, stream>>>(img, wsp);
    downsample_kernel<true><<<nblocks, TPB, 0, stream>>>(img, kern, offh, offv, wsp, out);
  } else {
    downsample_kernel<false><<<nblocks, TPB, 0, stream>>>(img, kern, offh, offv, nullptr, out);
  }
}